// GMEmbedding_47347719471275
// MI455X (gfx1250) — compile-verified
//
#include <hip/hip_runtime.h>

typedef float v2f __attribute__((ext_vector_type(2)));
typedef float v8f __attribute__((ext_vector_type(8)));

#define KK   512
#define DD   128
#define NN   65536
#define HWC  4096
#define LOGNORM (-0.9189385332046727f)   // -0.5*log(2*pi)

__device__ __forceinline__ v8f v8zero() {
  v8f z = {0.f,0.f,0.f,0.f,0.f,0.f,0.f,0.f};
  return z;
}

// ---------------------------------------------------------------------------
// prep: build A-matrix G[k][0:128] = -0.5*inv_var, G[k][128:256] = mu*inv_var,
//       constk[k] = -0.5*(sum logvar + sum mu^2 inv_var) + D*LOGNORM.
//       Also zeroes the moment accumulators for this iteration.
// ---------------------------------------------------------------------------
__global__ __launch_bounds__(128) void prep_kernel(
    const float* __restrict__ mu, const float* __restrict__ lv,
    float* __restrict__ G, float* __restrict__ constk,
    float* __restrict__ Px, float* __restrict__ Px2, float* __restrict__ Nk)
{
  __shared__ float red[128];
  const int k = blockIdx.x, d = threadIdx.x;
  const float l = lv[k*DD + d];
  const float m = mu[k*DD + d];
  const float iv = __expf(-l);
  G[k*256 + d]      = -0.5f * iv;
  G[k*256 + DD + d] = m * iv;
  Px[k*DD + d]  = 0.f;
  Px2[k*DD + d] = 0.f;
  if (d == 0) Nk[k] = 0.f;
  red[d] = l + m*m*iv;
  for (int s = 64; s > 0; s >>= 1) {
    __syncthreads();
    if (d < s) red[d] += red[d + s];
  }
  if (d == 0) constk[k] = -0.5f * red[0] + DD * LOGNORM;
}

// ---------------------------------------------------------------------------
// ll kernel: block = 8 waves, owns ALL K=512 rows x 32 columns of ll.
//   GEMM: ll[k,n] = sum_{c<256} G[k,c] * Bp[c,n]   (Bp = concat(x^2, x))
//   MODE 0: softmax over k, write P[k,n] to Pbuf ([K,N] row-major)
//   MODE 1: argmax over k (ties -> lowest k), write idx[n]
// ---------------------------------------------------------------------------
#define BPS 40   // LDS row stride: b128-aligned staging + lo/hi bank ranges disjoint
template<int MODE>
__global__ __launch_bounds__(256) void ll_kernel(
    const float* __restrict__ x, const float* __restrict__ G,
    const float* __restrict__ constk,
    float* __restrict__ Pbuf, int* __restrict__ idxOut)
{
  __shared__ float Bp[256 * BPS];
  __shared__ float redA[8][32];
  __shared__ float redB[8][32];
  __shared__ int   redI[8][32];

  const int tid  = threadIdx.x;
  const int lane = tid & 31;
  const int wave = tid >> 5;
  const int n0   = blockIdx.x * 32;
  const int b    = n0 >> 12;          // / 4096  (tiles never cross a batch)
  const int hw0  = n0 & (HWC - 1);

  { // stage B panel: rows [0,128) = x^2 ; rows [128,256) = x   (both [c][n])
    const int d    = tid >> 1;
    const int half = tid & 1;
    const float4* src = (const float4*)(x + ((size_t)b*DD + d)*HWC + hw0 + half*16);
    float4* d2 = (float4*)(Bp + d*BPS + half*16);
    float4* d1 = (float4*)(Bp + (DD + d)*BPS + half*16);
#pragma unroll
    for (int j = 0; j < 4; ++j) {
      float4 v = src[j];
      d1[j] = v;
      d2[j] = make_float4(v.x*v.x, v.y*v.y, v.z*v.z, v.w*v.w);
    }
  }
  __syncthreads();

  const int k0 = wave * 64;           // this wave's 64 rows of K
  const int lo = lane & 15;
  const int hi = lane >> 4;

  v8f acc[4][2];
#pragma unroll
  for (int mt = 0; mt < 4; ++mt) { acc[mt][0] = v8zero(); acc[mt][1] = v8zero(); }

#pragma unroll 2
  for (int c0 = 0; c0 < 256; c0 += 4) {
    const int cr = c0 + hi*2;
    // B tiles 4x16: VGPR0 = rows {c0, c0+2}, VGPR1 = rows {c0+1, c0+3}
    v2f bt0, bt1;
    bt0.x = Bp[cr*BPS + lo];       bt0.y = Bp[(cr+1)*BPS + lo];
    bt1.x = Bp[cr*BPS + 16 + lo];  bt1.y = Bp[(cr+1)*BPS + 16 + lo];
#pragma unroll
    for (int mt = 0; mt < 4; ++mt) {
      // A tile 16x4: lane = row, VGPR0 = cols {c0, c0+2}, VGPR1 = {c0+1, c0+3}
      v2f at = *(const v2f*)(G + (size_t)(k0 + mt*16 + lo)*256 + cr);
      acc[mt][0] = __builtin_amdgcn_wmma_f32_16x16x4_f32(false, at, false, bt0,
                      (short)0, acc[mt][0], false, false);
      acc[mt][1] = __builtin_amdgcn_wmma_f32_16x16x4_f32(false, at, false, bt1,
                      (short)0, acc[mt][1], false, false);
    }
  }

  // add per-row constant (C/D layout: VGPR r -> M = r + hi*8, N = lo)
#pragma unroll
  for (int mt = 0; mt < 4; ++mt)
#pragma unroll
    for (int r = 0; r < 8; ++r) {
      const float ck = constk[k0 + mt*16 + hi*8 + r];
      acc[mt][0][r] += ck;
      acc[mt][1][r] += ck;
    }

  if (MODE == 0) {
    // -------- softmax over k (512 rows, across 8 waves) --------
    float mx[2] = {-3.4e38f, -3.4e38f};
#pragma unroll
    for (int mt = 0; mt < 4; ++mt)
#pragma unroll
      for (int r = 0; r < 8; ++r) {
        mx[0] = fmaxf(mx[0], acc[mt][0][r]);
        mx[1] = fmaxf(mx[1], acc[mt][1][r]);
      }
#pragma unroll
    for (int nt = 0; nt < 2; ++nt)
      mx[nt] = fmaxf(mx[nt], __shfl_xor(mx[nt], 16, 32));
    if (lane < 16) { redA[wave][lo] = mx[0]; redA[wave][16 + lo] = mx[1]; }
    __syncthreads();
    float cmax[2];
#pragma unroll
    for (int nt = 0; nt < 2; ++nt) {
      float m = -3.4e38f;
#pragma unroll
      for (int w = 0; w < 8; ++w) m = fmaxf(m, redA[w][nt*16 + lo]);
      cmax[nt] = m;
    }
    float sm[2] = {0.f, 0.f};
#pragma unroll
    for (int mt = 0; mt < 4; ++mt)
#pragma unroll
      for (int r = 0; r < 8; ++r) {
        float e0 = __expf(acc[mt][0][r] - cmax[0]);
        float e1 = __expf(acc[mt][1][r] - cmax[1]);
        acc[mt][0][r] = e0; acc[mt][1][r] = e1;
        sm[0] += e0; sm[1] += e1;
      }
#pragma unroll
    for (int nt = 0; nt < 2; ++nt)
      sm[nt] += __shfl_xor(sm[nt], 16, 32);
    if (lane < 16) { redB[wave][lo] = sm[0]; redB[wave][16 + lo] = sm[1]; }
    __syncthreads();
    float inv[2];
#pragma unroll
    for (int nt = 0; nt < 2; ++nt) {
      float t = 0.f;
#pragma unroll
      for (int w = 0; w < 8; ++w) t += redB[w][nt*16 + lo];
      inv[nt] = 1.0f / t;
    }
#pragma unroll
    for (int mt = 0; mt < 4; ++mt)
#pragma unroll
      for (int r = 0; r < 8; ++r) {
        const int row = k0 + mt*16 + hi*8 + r;
        Pbuf[(size_t)row*NN + n0 + lo]      = acc[mt][0][r] * inv[0];
        Pbuf[(size_t)row*NN + n0 + 16 + lo] = acc[mt][1][r] * inv[1];
      }
  } else {
    // -------- argmax over k (first-max like jnp.argmax) --------
    float bv[2] = {-3.4e38f, -3.4e38f};
    int   bi[2] = {0x7fffffff, 0x7fffffff};
#pragma unroll
    for (int mt = 0; mt < 4; ++mt)
#pragma unroll
      for (int r = 0; r < 8; ++r) {   // rows scanned ascending within lane
        const int row = k0 + mt*16 + hi*8 + r;
#pragma unroll
        for (int nt = 0; nt < 2; ++nt) {
          const float v = acc[mt][nt][r];
          if (v > bv[nt]) { bv[nt] = v; bi[nt] = row; }
        }
      }
#pragma unroll
    for (int nt = 0; nt < 2; ++nt) {
      const float ov = __shfl_xor(bv[nt], 16, 32);
      const int   oi = __shfl_xor(bi[nt], 16, 32);
      if (ov > bv[nt] || (ov == bv[nt] && oi < bi[nt])) { bv[nt] = ov; bi[nt] = oi; }
    }
    if (lane < 16) {
      redA[wave][lo]      = bv[0]; redI[wave][lo]      = bi[0];
      redA[wave][16 + lo] = bv[1]; redI[wave][16 + lo] = bi[1];
    }
    __syncthreads();
    if (tid < 32) {
      float v = -3.4e38f; int i = 0x7fffffff;
#pragma unroll
      for (int w = 0; w < 8; ++w) {
        const float vv = redA[w][tid]; const int ii = redI[w][tid];
        if (vv > v || (vv == v && ii < i)) { v = vv; i = ii; }
      }
      idxOut[n0 + tid] = i;
    }
  }
}

// ---------------------------------------------------------------------------
// moments: Px = P @ x, Px2 = P @ x^2, Nk = rowsum(P).
// grid = 32 n-slabs (2048 cols) x 16 k-tiles of 32 rows = 512 blocks.
// Software pipeline: next chunk global->regs while WMMA on current LDS buffer,
// regs->other LDS buffer, one barrier per chunk. x kept in native [d][n]
// layout (no transpose): b128 staging stores + single b64 B-tile reads.
// ---------------------------------------------------------------------------
#define MCH  32   // chunk columns
#define XDS  36   // [d][n] stride: b128-aligned stores, conflict-free b64 reads
#define PCS  36   // P chunk stride
__global__ __launch_bounds__(256) void moments_kernel(
    const float* __restrict__ x, const float* __restrict__ Pbuf,
    float* __restrict__ Px, float* __restrict__ Px2, float* __restrict__ Nk)
{
  __shared__ float XD[2][DD * XDS];   // 2 x 18.4 KB
  __shared__ float PC[2][32 * PCS];   // 2 x 4.6 KB

  const int tid  = threadIdx.x;
  const int lane = tid & 31;
  const int wave = tid >> 5;
  const int slab = blockIdx.x >> 4;        // 0..31, 2048 columns each
  const int kt0  = (blockIdx.x & 15) * 32; // k-tile base
  const int b      = slab >> 1;            // batch image
  const int nbase  = slab * 2048;          // global column base
  const int hwbase = (slab & 1) * 2048;    // offset inside image
  const int dbase = wave * 16;
  const int lo = lane & 15;
  const int hi = lane >> 4;

  // staging assignments
  const int prow = tid >> 3;               // 0..31 : P row
  const int pcol = (tid & 7) * 4;          // 4 P cols / thread
  const int xd   = tid >> 1;               // 0..127 : x row (d)
  const int xh   = (tid & 1) * 16;         // 16 x cols / thread

  const float* Pbase = Pbuf + (size_t)(kt0 + prow)*NN + nbase + pcol;
  const float* Xbase = x + ((size_t)b*DD + xd)*HWC + hwbase + xh;

  v8f cpx[2], cq[2];
  cpx[0] = v8zero(); cpx[1] = v8zero();
  cq[0]  = v8zero(); cq[1]  = v8zero();
  float nkp = 0.f;

  // ---- prologue: stage chunk 0 ----
  {
    float4 pr  = *(const float4*)Pbase;
    const float4* xs = (const float4*)Xbase;
    float4 x0 = xs[0], x1 = xs[1], x2 = xs[2], x3 = xs[3];
    *(float4*)(&PC[0][prow*PCS + pcol]) = pr;
    nkp += pr.x + pr.y + pr.z + pr.w;
    float4* xdst = (float4*)(&XD[0][xd*XDS + xh]);
    xdst[0] = x0; xdst[1] = x1; xdst[2] = x2; xdst[3] = x3;
  }
  __syncthreads();

  for (int ch = 0; ch < 64; ++ch) {
    const int cur = ch & 1;
    const int nxt = cur ^ 1;

    // issue next chunk's global loads before compute (latency overlap)
    float4 prn, xn0, xn1, xn2, xn3;
    if (ch < 63) {
      const int off = (ch + 1) * MCH;
      prn = *(const float4*)(Pbase + off);
      const float4* xs = (const float4*)(Xbase + off);
      xn0 = xs[0]; xn1 = xs[1]; xn2 = xs[2]; xn3 = xs[3];
    }

    // WMMA over current chunk (reduce 32 n in steps of 4)
#pragma unroll
    for (int c0 = 0; c0 < MCH; c0 += 4) {
      const int nr = c0 + hi*2;
      // B tile 4x16 (rows n, cols d) from native-[d][n] tile: one b64 read
      v2f bt = *(const v2f*)(&XD[cur][(dbase + lo)*XDS + nr]);
      v2f b2; b2.x = bt.x*bt.x; b2.y = bt.y*bt.y;
#pragma unroll
      for (int mt = 0; mt < 2; ++mt) {
        v2f at = *(const v2f*)(&PC[cur][(mt*16 + lo)*PCS + nr]);
        cpx[mt] = __builtin_amdgcn_wmma_f32_16x16x4_f32(false, at, false, bt,
                      (short)0, cpx[mt], false, false);
        cq[mt]  = __builtin_amdgcn_wmma_f32_16x16x4_f32(false, at, false, b2,
                      (short)0, cq[mt], false, false);
      }
    }

    // drain next chunk into the other buffer
    if (ch < 63) {
      *(float4*)(&PC[nxt][prow*PCS + pcol]) = prn;
      nkp += prn.x + prn.y + prn.z + prn.w;
      float4* xdst = (float4*)(&XD[nxt][xd*XDS + xh]);
      xdst[0] = xn0; xdst[1] = xn1; xdst[2] = xn2; xdst[3] = xn3;
    }
    __syncthreads();
  }

#pragma unroll
  for (int mt = 0; mt < 2; ++mt)
#pragma unroll
    for (int r = 0; r < 8; ++r) {
      const int row = kt0 + mt*16 + hi*8 + r;
      atomicAdd(&Px [row*DD + dbase + lo], cpx[mt][r]);
      atomicAdd(&Px2[row*DD + dbase + lo], cq[mt][r]);
    }
  atomicAdd(&Nk[kt0 + prow], nkp);
}

// ---------------------------------------------------------------------------
__global__ __launch_bounds__(128) void update_kernel(
    const float* __restrict__ Px, const float* __restrict__ Px2,
    const float* __restrict__ Nk, float* __restrict__ mu, float* __restrict__ lv)
{
  const int k = blockIdx.x, d = threadIdx.x;
  const float nk = Nk[k];
  const float denom = nk + 1e-6f;
  const float px = Px[k*DD + d], px2 = Px2[k*DD + d];
  const float m = px / denom;
  const float var = (px2 - 2.f*m*px + m*m*nk) / denom;
  mu[k*DD + d] = m;
  lv[k*DD + d] = __logf(fmaxf(var, 1e-6f));
}

__global__ __launch_bounds__(128) void newmu_kernel(
    const float* __restrict__ emu, const float* __restrict__ mu,
    float* __restrict__ nmu)
{
  const int i = blockIdx.x*128 + threadIdx.x;
  nmu[i] = 0.9f*emu[i] + 0.1f*mu[i];
}

__global__ __launch_bounds__(256) void enc_kernel(
    const int* __restrict__ idx, float* __restrict__ enc)
{
  const int n = blockIdx.x;
  const int i = idx[n];
  const int c0 = threadIdx.x * 2;
  float2 v;
  v.x = (c0     == i) ? 1.f : 0.f;
  v.y = (c0 + 1 == i) ? 1.f : 0.f;
  *(float2*)(enc + (size_t)n*KK + c0) = v;
}

__global__ __launch_bounds__(256) void quant_kernel(
    const int* __restrict__ idx, const float* __restrict__ nmu,
    float* __restrict__ quant)
{
  const int bd = blockIdx.x;          // b*128 + d
  const int b = bd >> 7, d = bd & 127;
  const int* ip = idx + b*HWC;
  float* qp = quant + (size_t)bd * HWC;
  for (int j = threadIdx.x; j < HWC; j += blockDim.x)
    qp[j] = nmu[ip[j]*DD + d];
}

// ---------------------------------------------------------------------------
extern "C" void kernel_launch(void* const* d_in, const int* in_sizes, int n_in,
                              void* d_out, int out_size, void* d_ws, size_t ws_size,
                              hipStream_t stream)
{
  const float* x      = (const float*)d_in[0];
  const float* emb_mu = (const float*)d_in[1];
  // d_in[2]=emb_logvar, d_in[3]=emb_pi: state-only in reference, not in outputs
  const float* mu0    = (const float*)d_in[4];
  const float* lv0    = (const float*)d_in[5];

  float* enc   = (float*)d_out;                  // [N, K]
  float* quant = enc + (size_t)NN * KK;          // [B, D, H, W]
  float* Pbuf  = enc;                            // reuse enc region as P [K, N]

  float* ws = (float*)d_ws;
  float* G      = ws;  ws += KK*256;
  float* constk = ws;  ws += KK;
  float* muW    = ws;  ws += KK*DD;
  float* lvW    = ws;  ws += KK*DD;
  float* PxW    = ws;  ws += KK*DD;
  float* Px2W   = ws;  ws += KK*DD;
  float* NkW    = ws;  ws += KK;
  float* nmuW   = ws;  ws += KK*DD;
  int*   idxW   = (int*)ws;

  hipMemcpyAsync(muW, mu0, (size_t)KK*DD*sizeof(float), hipMemcpyDeviceToDevice, stream);
  hipMemcpyAsync(lvW, lv0, (size_t)KK*DD*sizeof(float), hipMemcpyDeviceToDevice, stream);

  for (int it = 0; it < 3; ++it) {
    prep_kernel<<<KK, 128, 0, stream>>>(muW, lvW, G, constk, PxW, Px2W, NkW);
    ll_kernel<0><<<NN/32, 256, 0, stream>>>(x, G, constk, Pbuf, nullptr);
    moments_kernel<<<512, 256, 0, stream>>>(x, Pbuf, PxW, Px2W, NkW);
    update_kernel<<<KK, 128, 0, stream>>>(PxW, Px2W, NkW, muW, lvW);
  }
  prep_kernel<<<KK, 128, 0, stream>>>(muW, lvW, G, constk, PxW, Px2W, NkW);
  ll_kernel<1><<<NN/32, 256, 0, stream>>>(x, G, constk, Pbuf, idxW);
  newmu_kernel<<<KK, 128, 0, stream>>>(emb_mu, muW, nmuW);
  enc_kernel<<<NN, 256, 0, stream>>>(idxW, enc);
  quant_kernel<<<16*DD, 256, 0, stream>>>(idxW, nmuW, quant);

  (void)in_sizes; (void)n_in; (void)out_size; (void)ws_size;
}